// TSFN_Model_429496729784
// MI455X (gfx1250) — compile-verified
//
#include <hip/hip_runtime.h>
#include <hip/hip_bf16.h>
#include <math.h>

// ---------------- model constants ----------------
#define NB   512   // batch (graph nodes)
#define TT   10    // time steps
#define HH   40
#define WW   40
#define EE   8192  // edges
#define FEAT 616   // 400 (veg) + 200 (cwsi) + 16 (emb)

typedef float v2f __attribute__((ext_vector_type(2)));
typedef float v8f __attribute__((ext_vector_type(8)));

// ---------------- async LDS staging (gfx1250 GLOBAL_LOAD_ASYNC_TO_LDS) -------
#if __has_builtin(__builtin_amdgcn_global_load_async_to_lds_b128)
#define HAS_ASYNC 1
typedef int v4i __attribute__((__vector_size__(16)));
typedef __attribute__((address_space(1))) v4i gv4i;   // global 16B vector
typedef __attribute__((address_space(3))) v4i lv4i;   // LDS 16B vector
#endif

__device__ __forceinline__ void lds_copy16(void* lds, const void* g) {
#ifdef HAS_ASYNC
  __builtin_amdgcn_global_load_async_to_lds_b128(
      (gv4i*)(uintptr_t)g, (lv4i*)(unsigned)(uintptr_t)lds, 0, 0);
#else
  *(float4*)lds = *(const float4*)g;
#endif
}
__device__ __forceinline__ void lds_copy_wait() {
#ifdef HAS_ASYNC
#if __has_builtin(__builtin_amdgcn_s_wait_asynccnt)
  __builtin_amdgcn_s_wait_asynccnt(0);
#else
  asm volatile("s_wait_asynccnt 0" ::: "memory");
#endif
#endif
}
// stage n16 16-byte chunks from gSrc to ldsDst (both 16B aligned)
__device__ __forceinline__ void stage_lds(float* ldsDst, const float* gSrc,
                                          int n16, int tid) {
  for (int i = tid; i < n16; i += 256)
    lds_copy16((char*)ldsDst + (size_t)i * 16, (const char*)gSrc + (size_t)i * 16);
}

// ---------------- WMMA helper (fp32 16x16x4) ----------------
__device__ __forceinline__ v8f wmma_f32(v2f a, v2f b, v8f c) {
  // 8 args: (neg_a, A, neg_b, B, c_mod, C, reuse_a, reuse_b)
  return __builtin_amdgcn_wmma_f32_16x16x4_f32(false, a, false, b, (short)0, c,
                                               false, false);
}

__device__ __forceinline__ float sigf(float x) { return 1.0f / (1.0f + expf(-x)); }

// order-preserving float<->uint encoding for atomic max on floats
__device__ __forceinline__ unsigned encOrd(float f) {
  unsigned b = __float_as_uint(f);
  return (b & 0x80000000u) ? ~b : (b | 0x80000000u);
}
__device__ __forceinline__ float decOrdZero(unsigned u) {
  unsigned b = (u & 0x80000000u) ? (u & 0x7fffffffu) : ~u;
  float f = __uint_as_float(b);
  // -inf (no incoming edge) -> 0, matching jnp.where(isfinite(agg), agg, 0)
  return (f == -__builtin_huge_valf()) ? 0.0f : f;
}
#define ENC_NEG_INF 0x007FFFFFu  // encOrd(-inf)

// ---------------- small utility kernels ----------------
__global__ void fill_u32_kernel(unsigned* p, unsigned v, int n) {
  int i = blockIdx.x * blockDim.x + threadIdx.x;
  if (i < n) p[i] = v;
}

// stats[t][c][2] = {sum, sumsq} ; produce scsh[t][c][2] = {scale, shift}
__global__ void bn_finalize_kernel(const float* __restrict__ stats,
                                   const float* __restrict__ g,
                                   const float* __restrict__ be,
                                   float* __restrict__ scsh, int TC, int C,
                                   float invCnt) {
  int idx = blockIdx.x * blockDim.x + threadIdx.x;
  if (idx >= TC) return;
  int c = idx % C;
  float s0 = stats[idx * 2 + 0];
  float s1 = stats[idx * 2 + 1];
  float mean = s0 * invCnt;
  float var = s1 * invCnt - mean * mean;
  float scale = g[c] * rsqrtf(var + 1e-5f);
  scsh[idx * 2 + 0] = scale;
  scsh[idx * 2 + 1] = be[c] - scale * mean;
}

// ---------------- conv helpers ----------------
template <int CIN>
__device__ __forceinline__ float conv1_at(const float* __restrict__ s_in,
                                          const float* __restrict__ wc,
                                          float bias, int y, int x) {
  float acc = bias;
#pragma unroll
  for (int ci = 0; ci < CIN; ++ci) {
    const float* ip = s_in + ci * 1600;
    const float* wp = wc + ci * 9;
#pragma unroll
    for (int ky = 0; ky < 3; ++ky) {
      int yy = y + ky - 1;
      if ((unsigned)yy < 40u) {
#pragma unroll
        for (int kx = 0; kx < 3; ++kx) {
          int xx = x + kx - 1;
          if ((unsigned)xx < 40u) acc += ip[yy * 40 + xx] * wp[ky * 3 + kx];
        }
      }
    }
  }
  return acc;
}

// conv2 3x3 SAME on 20x20 pooled maps (16 in channels), weights from global
__device__ __forceinline__ float conv2_at(const float* __restrict__ s_pool,
                                          const float* __restrict__ w2c,
                                          float bias, int py, int px) {
  float acc = bias;
#pragma unroll
  for (int ci = 0; ci < 16; ++ci) {
    const float* ip = s_pool + ci * 400;
    const float* wp = w2c + ci * 9;
#pragma unroll
    for (int ky = 0; ky < 3; ++ky) {
      int yy = py + ky - 1;
      if ((unsigned)yy < 20u) {
#pragma unroll
        for (int kx = 0; kx < 3; ++kx) {
          int xx = px + kx - 1;
          if ((unsigned)xx < 20u) acc += ip[yy * 20 + xx] * wp[ky * 3 + kx];
        }
      }
    }
  }
  return acc;
}

// ---------------- pass 1: conv1 + BN1 statistics ----------------
template <int CIN>
__global__ __launch_bounds__(256) void conv1_stats_kernel(
    const float* __restrict__ in, const float* __restrict__ w1,
    const float* __restrict__ b1, float* __restrict__ stats /*[T][16][2]*/) {
  __shared__ __align__(16) float s_in[CIN * 1600];
  __shared__ __align__(16) float s_w[16 * CIN * 9];
  __shared__ float s_b[16];
  __shared__ float s_stat[32];

  const int tid = threadIdx.x;
  const int img = blockIdx.x;          // n*T + t
  const int t = img % TT;

  stage_lds(s_in, in + (size_t)img * CIN * 1600, CIN * 400, tid);
  stage_lds(s_w, w1, 16 * CIN * 9 / 4, tid);
  if (tid < 16) s_b[tid] = b1[tid];
  if (tid < 32) s_stat[tid] = 0.0f;
  lds_copy_wait();
  __syncthreads();

  float ls[16], lss[16];
#pragma unroll
  for (int c = 0; c < 16; ++c) { ls[c] = 0.0f; lss[c] = 0.0f; }

  for (int p = tid; p < 1600; p += 256) {
    int y = p / 40, x = p % 40;
#pragma unroll
    for (int c = 0; c < 16; ++c) {
      float v = conv1_at<CIN>(s_in, s_w + c * CIN * 9, s_b[c], y, x);
      ls[c] += v;
      lss[c] += v * v;
    }
  }
#pragma unroll
  for (int c = 0; c < 16; ++c) {
    atomicAdd(&s_stat[c * 2 + 0], ls[c]);
    atomicAdd(&s_stat[c * 2 + 1], lss[c]);
  }
  __syncthreads();
  if (tid < 32) atomicAdd(&stats[t * 32 + tid], s_stat[tid]);
}

// shared pool stage: conv1 -> BN1 affine (maxpool commuted via scale sign) -> relu -> 20x20
template <int CIN>
__device__ __forceinline__ void make_pool(const float* __restrict__ s_in,
                                          const float* __restrict__ s_w1,
                                          const float* __restrict__ s_b1,
                                          const float* __restrict__ s_a1,
                                          const float* __restrict__ s_s1,
                                          float* __restrict__ s_pool, int tid) {
  for (int idx = tid; idx < 16 * 400; idx += 256) {
    int c = idx / 400, p = idx % 400;
    int py = p / 20, px = p % 20;
    const float* wc = s_w1 + c * CIN * 9;
    float v00 = conv1_at<CIN>(s_in, wc, s_b1[c], 2 * py + 0, 2 * px + 0);
    float v01 = conv1_at<CIN>(s_in, wc, s_b1[c], 2 * py + 0, 2 * px + 1);
    float v10 = conv1_at<CIN>(s_in, wc, s_b1[c], 2 * py + 1, 2 * px + 0);
    float v11 = conv1_at<CIN>(s_in, wc, s_b1[c], 2 * py + 1, 2 * px + 1);
    float sc = s_a1[c];
    float raw;
    if (sc >= 0.0f)
      raw = fmaxf(fmaxf(v00, v01), fmaxf(v10, v11));
    else
      raw = fminf(fminf(v00, v01), fminf(v10, v11));
    float val = fmaxf(sc * raw + s_s1[c], 0.0f);
    s_pool[c * 400 + p] = val;
  }
}

// ---------------- pass 2: conv2 statistics (recompute chain) ----------------
template <int CIN, int C2>
__global__ __launch_bounds__(256) void conv2_stats_kernel(
    const float* __restrict__ in, const float* __restrict__ w1,
    const float* __restrict__ b1, const float* __restrict__ scsh1,
    const float* __restrict__ w2, const float* __restrict__ b2,
    float* __restrict__ stats2 /*[T][C2][2]*/) {
  __shared__ __align__(16) float s_in[CIN * 1600];
  __shared__ __align__(16) float s_pool[16 * 400];
  __shared__ __align__(16) float s_w1[16 * CIN * 9];
  __shared__ float s_b1[16], s_a1[16], s_s1[16];
  __shared__ float s_stat[C2 * 2];

  const int tid = threadIdx.x;
  const int img = blockIdx.x;
  const int t = img % TT;

  stage_lds(s_in, in + (size_t)img * CIN * 1600, CIN * 400, tid);
  stage_lds(s_w1, w1, 16 * CIN * 9 / 4, tid);
  if (tid < 16) {
    s_b1[tid] = b1[tid];
    s_a1[tid] = scsh1[(t * 16 + tid) * 2 + 0];
    s_s1[tid] = scsh1[(t * 16 + tid) * 2 + 1];
  }
  if (tid < C2 * 2) s_stat[tid] = 0.0f;
  lds_copy_wait();
  __syncthreads();

  make_pool<CIN>(s_in, s_w1, s_b1, s_a1, s_s1, s_pool, tid);
  __syncthreads();

#pragma unroll
  for (int c2 = 0; c2 < C2; ++c2) {
    float ls = 0.0f, lss = 0.0f;
    const float* w2c = w2 + c2 * 16 * 9;
    float bias = b2[c2];
    for (int p = tid; p < 400; p += 256) {
      float v = conv2_at(s_pool, w2c, bias, p / 20, p % 20);
      ls += v;
      lss += v * v;
    }
    atomicAdd(&s_stat[c2 * 2 + 0], ls);
    atomicAdd(&s_stat[c2 * 2 + 1], lss);
  }
  __syncthreads();
  if (tid < C2 * 2) atomicAdd(&stats2[t * C2 * 2 + tid], s_stat[tid]);
}

// ---------------- pass 3: full features (conv1->pool->conv2->BN2->relu->avgpool) ----
template <int CIN, int C2>
__global__ __launch_bounds__(256) void features_kernel(
    const float* __restrict__ in, const float* __restrict__ w1,
    const float* __restrict__ b1, const float* __restrict__ scsh1,
    const float* __restrict__ w2, const float* __restrict__ b2,
    const float* __restrict__ scsh2, float* __restrict__ feats, int foff) {
  constexpr int BUF = (CIN * 1600 > C2 * 400) ? CIN * 1600 : C2 * 400;
  __shared__ __align__(16) float s_buf[BUF];  // input, later conv2 output
  __shared__ __align__(16) float s_pool[16 * 400];
  __shared__ __align__(16) float s_w1[16 * CIN * 9];
  __shared__ float s_b1[16], s_a1[16], s_s1[16];
  __shared__ float s_a2[C2], s_s2[C2];

  const int tid = threadIdx.x;
  const int img = blockIdx.x;
  const int n = img / TT;
  const int t = img % TT;

  stage_lds(s_buf, in + (size_t)img * CIN * 1600, CIN * 400, tid);
  stage_lds(s_w1, w1, 16 * CIN * 9 / 4, tid);
  if (tid < 16) {
    s_b1[tid] = b1[tid];
    s_a1[tid] = scsh1[(t * 16 + tid) * 2 + 0];
    s_s1[tid] = scsh1[(t * 16 + tid) * 2 + 1];
  }
  if (tid < C2) {
    s_a2[tid] = scsh2[(t * C2 + tid) * 2 + 0];
    s_s2[tid] = scsh2[(t * C2 + tid) * 2 + 1];
  }
  lds_copy_wait();
  __syncthreads();

  make_pool<CIN>(s_buf, s_w1, s_b1, s_a1, s_s1, s_pool, tid);
  __syncthreads();  // s_buf free now

  // conv2 + BN2 + relu into s_buf
  for (int idx = tid; idx < C2 * 400; idx += 256) {
    int c2 = idx / 400, p = idx % 400;
    float v = conv2_at(s_pool, w2 + c2 * 16 * 9, b2[c2], p / 20, p % 20);
    s_buf[idx] = fmaxf(s_a2[c2] * v + s_s2[c2], 0.0f);
  }
  __syncthreads();

  // 4x4 average pool -> 5x5, write features
  for (int idx = tid; idx < C2 * 25; idx += 256) {
    int c2 = idx / 25, q = idx % 25;
    int i = q / 5, j = q % 5;
    float s = 0.0f;
#pragma unroll
    for (int dy = 0; dy < 4; ++dy)
#pragma unroll
      for (int dx = 0; dx < 4; ++dx)
        s += s_buf[c2 * 400 + (i * 4 + dy) * 20 + (j * 4 + dx)];
    feats[((size_t)t * NB + n) * FEAT + foff + idx] = s * (1.0f / 16.0f);
  }
}

// ---------------- embedding ----------------
__global__ void emb_kernel(const int* __restrict__ irr,
                           const float* __restrict__ emb,
                           float* __restrict__ feats) {
  int idx = blockIdx.x * blockDim.x + threadIdx.x;  // N*T*16
  if (idx >= NB * TT * 16) return;
  int n = idx / (TT * 16);
  int r = idx % (TT * 16);
  int t = r / 16, j = r % 16;
  feats[((size_t)t * NB + n) * FEAT + 600 + j] = emb[irr[n * TT + t] * 16 + j];
}

// ---------------- SAGE max scatter ----------------
__global__ __launch_bounds__(256) void agg_scatter_kernel(
    const float* __restrict__ x, const int* __restrict__ src,
    const int* __restrict__ dst, unsigned* __restrict__ agg, int F) {
  int bid = blockIdx.x;      // t * E + e
  int t = bid / EE;
  int e = bid % EE;
  int s = src[e], d = dst[e];
  const float* xs = x + ((size_t)t * NB + s) * F;
  unsigned* ad = agg + ((size_t)t * NB + d) * F;
  for (int f = threadIdx.x; f < F; f += 256) atomicMax(&ad[f], encOrd(xs[f]));
}

// ---------------- SAGE GEMM: out = relu(dec(agg)@wl^T + bl + x@wr^T) -----------
// WMMA fp32 16x16x4. Block = 256 threads = 8 waves = rowTiles x colTiles tiles.
// Per lane, the two fragment elements of a k-step are contiguous: one b64 load.
__global__ __launch_bounds__(256) void sage_gemm_kernel(
    const unsigned* __restrict__ agg, const float* __restrict__ x,
    const float* __restrict__ wl, const float* __restrict__ bl,
    const float* __restrict__ wr, float* __restrict__ out, int K, int O,
    int colTiles) {
  const int tid = threadIdx.x;
  const int lane = tid & 31;
  const int w = tid >> 5;
  const int rowTiles = 8 / colTiles;
  const int ct = w % colTiles;
  const int rt = w / colTiles;
  const int hi = lane >> 4;
  const int ln = lane & 15;

  const int m0 = blockIdx.x * rowTiles * 16 + rt * 16;
  const int m = m0 + ln;
  const int col = ct * 16 + ln;

  // each lane's 2 fragment elements start at k + 2*hi (8B-aligned, contiguous)
  const unsigned* aRow = agg + (size_t)m * K + 2 * hi;
  const float* xRow = x + (size_t)m * K + 2 * hi;
  const float* wlRow = wl + (size_t)col * K + 2 * hi;
  const float* wrRow = wr + (size_t)col * K + 2 * hi;

  v8f acc;
#pragma unroll
  for (int i = 0; i < 8; ++i) acc[i] = 0.0f;

#pragma unroll 2
  for (int k = 0; k < K; k += 4) {
    uint2 au = *(const uint2*)(aRow + k);
    v2f bL = *(const v2f*)(wlRow + k);
    v2f aX = *(const v2f*)(xRow + k);
    v2f bR = *(const v2f*)(wrRow + k);
    v2f aA;
    aA.x = decOrdZero(au.x);
    aA.y = decOrdZero(au.y);
    acc = wmma_f32(aA, bL, acc);
    acc = wmma_f32(aX, bR, acc);
  }
  const float bias = bl[col];
#pragma unroll
  for (int r = 0; r < 8; ++r) {
    int mr = m0 + r + hi * 8;
    out[(size_t)mr * O + col] = fmaxf(acc[r] + bias, 0.0f);
  }
}

// ---------------- LSTM step (one kernel per t) ----------------
// Block: 32 batch rows, 8 waves, each wave owns 4 of the 32 16x16 z-tiles.
__global__ __launch_bounds__(256) void lstm_step_kernel(
    const float* __restrict__ x,   // (NB, 32) slice for this t
    const float* __restrict__ wih, // (256, 32)
    const float* __restrict__ whh, // (256, 64)
    const float* __restrict__ bih, const float* __restrict__ bhh,
    float* __restrict__ h, float* __restrict__ c) {
  __shared__ float s_z[32 * 256];
  const int tid = threadIdx.x;
  const int lane = tid & 31;
  const int w = tid >> 5;
  const int hi = lane >> 4;
  const int ln = lane & 15;
  const int blockRow = blockIdx.x * 32;
  const int rt = (w >= 4) ? 1 : 0;
  const int baseCt = (w & 3) * 4;  // 4 consecutive column tiles per wave
  const int m = blockRow + rt * 16 + ln;

  const float* xRow = x + (size_t)m * 32 + 2 * hi;
  const float* hRow = h + (size_t)m * 64 + 2 * hi;

  v8f acc[4];
#pragma unroll
  for (int i = 0; i < 4; ++i)
#pragma unroll
    for (int r = 0; r < 8; ++r) acc[i][r] = 0.0f;

  // x @ wih^T   (K = 32)
#pragma unroll
  for (int k = 0; k < 32; k += 4) {
    v2f a = *(const v2f*)(xRow + k);
#pragma unroll
    for (int i = 0; i < 4; ++i) {
      int colg = (baseCt + i) * 16 + ln;
      v2f b = *(const v2f*)(wih + (size_t)colg * 32 + 2 * hi + k);
      acc[i] = wmma_f32(a, b, acc[i]);
    }
  }
  // h @ whh^T   (K = 64)
#pragma unroll
  for (int k = 0; k < 64; k += 4) {
    v2f a = *(const v2f*)(hRow + k);
#pragma unroll
    for (int i = 0; i < 4; ++i) {
      int colg = (baseCt + i) * 16 + ln;
      v2f b = *(const v2f*)(whh + (size_t)colg * 64 + 2 * hi + k);
      acc[i] = wmma_f32(a, b, acc[i]);
    }
  }
#pragma unroll
  for (int i = 0; i < 4; ++i) {
    int colg = (baseCt + i) * 16 + ln;
#pragma unroll
    for (int r = 0; r < 8; ++r) {
      int rl = rt * 16 + r + hi * 8;
      s_z[rl * 256 + colg] = acc[i][r];
    }
  }
  __syncthreads();

  // gates: i,f,g,o = split(z,4); c' = sig(f)*c + sig(i)*tanh(g); h' = sig(o)*tanh(c')
  for (int idx = tid; idx < 32 * 64; idx += 256) {
    int rl = idx >> 6;
    int j = idx & 63;
    int row = blockRow + rl;
    float zi = s_z[rl * 256 + j] + bih[j] + bhh[j];
    float zf = s_z[rl * 256 + 64 + j] + bih[64 + j] + bhh[64 + j];
    float zg = s_z[rl * 256 + 128 + j] + bih[128 + j] + bhh[128 + j];
    float zo = s_z[rl * 256 + 192 + j] + bih[192 + j] + bhh[192 + j];
    float cv = sigf(zf) * c[(size_t)row * 64 + j] + sigf(zi) * tanhf(zg);
    float hv = sigf(zo) * tanhf(cv);
    c[(size_t)row * 64 + j] = cv;
    h[(size_t)row * 64 + j] = hv;
  }
}

// ---------------- final FC ----------------
__global__ void fc_kernel(const float* __restrict__ h,
                          const float* __restrict__ fcw,
                          const float* __restrict__ fcb,
                          float* __restrict__ out) {
  int n = blockIdx.x * blockDim.x + threadIdx.x;
  if (n >= NB) return;
  float a = fcb[0];
#pragma unroll
  for (int j = 0; j < 64; ++j) a += h[(size_t)n * 64 + j] * fcw[j];
  out[n] = a;
}

// ---------------- host launcher ----------------
static inline char* carve(char*& p, size_t bytes) {
  char* r = p;
  p += (bytes + 255) & ~((size_t)255);
  return r;
}

extern "C" void kernel_launch(void* const* d_in, const int* in_sizes, int n_in,
                              void* d_out, int out_size, void* d_ws,
                              size_t ws_size, hipStream_t stream) {
  (void)in_sizes; (void)n_in; (void)out_size; (void)ws_size;
  const float* veg  = (const float*)d_in[0];
  const float* cwsi = (const float*)d_in[1];
  const int*   irr  = (const int*)d_in[2];
  const int*   eidx = (const int*)d_in[3];
  const float* vw1 = (const float*)d_in[4];
  const float* vb1 = (const float*)d_in[5];
  const float* vg1 = (const float*)d_in[6];
  const float* vbe1 = (const float*)d_in[7];
  const float* vw2 = (const float*)d_in[8];
  const float* vb2 = (const float*)d_in[9];
  const float* vg2 = (const float*)d_in[10];
  const float* vbe2 = (const float*)d_in[11];
  const float* cw1 = (const float*)d_in[12];
  const float* cb1 = (const float*)d_in[13];
  const float* cg1 = (const float*)d_in[14];
  const float* cbe1 = (const float*)d_in[15];
  const float* cw2 = (const float*)d_in[16];
  const float* cb2 = (const float*)d_in[17];
  const float* cg2 = (const float*)d_in[18];
  const float* cbe2 = (const float*)d_in[19];
  const float* emb = (const float*)d_in[20];
  const float* s1_wl = (const float*)d_in[21];
  const float* s1_bl = (const float*)d_in[22];
  const float* s1_wr = (const float*)d_in[23];
  const float* s2_wl = (const float*)d_in[24];
  const float* s2_bl = (const float*)d_in[25];
  const float* s2_wr = (const float*)d_in[26];
  const float* wih = (const float*)d_in[27];
  const float* whh = (const float*)d_in[28];
  const float* bih = (const float*)d_in[29];
  const float* bhh = (const float*)d_in[30];
  const float* fcw = (const float*)d_in[31];
  const float* fcb = (const float*)d_in[32];
  float* out = (float*)d_out;

  const int* src = eidx;
  const int* dst = eidx + EE;

  // workspace carve
  char* p = (char*)d_ws;
  float* statsBase = (float*)p;  // all stats zeroed in one memset
  float* st1v = (float*)carve(p, TT * 16 * 2 * sizeof(float));
  float* st2v = (float*)carve(p, TT * 16 * 2 * sizeof(float));
  float* st1c = (float*)carve(p, TT * 16 * 2 * sizeof(float));
  float* st2c = (float*)carve(p, TT * 8 * 2 * sizeof(float));
  size_t statsBytes = (size_t)((char*)p - (char*)statsBase);
  float* sc1v = (float*)carve(p, TT * 16 * 2 * sizeof(float));
  float* sc2v = (float*)carve(p, TT * 16 * 2 * sizeof(float));
  float* sc1c = (float*)carve(p, TT * 16 * 2 * sizeof(float));
  float* sc2c = (float*)carve(p, TT * 8 * 2 * sizeof(float));
  float* feats = (float*)carve(p, (size_t)TT * NB * FEAT * sizeof(float));
  unsigned* agg1 = (unsigned*)carve(p, (size_t)TT * NB * FEAT * sizeof(unsigned));
  float* h1 = (float*)carve(p, (size_t)TT * NB * 64 * sizeof(float));
  unsigned* agg2 = (unsigned*)carve(p, (size_t)TT * NB * 64 * sizeof(unsigned));
  float* h2 = (float*)carve(p, (size_t)TT * NB * 32 * sizeof(float));
  float* hS = (float*)carve(p, (size_t)NB * 64 * sizeof(float));
  float* cS = (float*)carve(p, (size_t)NB * 64 * sizeof(float));

  (void)hipMemsetAsync(statsBase, 0, statsBytes, stream);
  (void)hipMemsetAsync(hS, 0, (size_t)NB * 64 * sizeof(float), stream);
  (void)hipMemsetAsync(cS, 0, (size_t)NB * 64 * sizeof(float), stream);

  const int nImg = NB * TT;  // 5120

  // ---- CNN branch: veg (Cin=5 -> 16 -> 16) ----
  conv1_stats_kernel<5><<<nImg, 256, 0, stream>>>(veg, vw1, vb1, st1v);
  // ---- CNN branch: cwsi (Cin=1 -> 16 -> 8) ----
  conv1_stats_kernel<1><<<nImg, 256, 0, stream>>>(cwsi, cw1, cb1, st1c);

  bn_finalize_kernel<<<(TT * 16 + 63) / 64, 64, 0, stream>>>(
      st1v, vg1, vbe1, sc1v, TT * 16, 16, 1.0f / (NB * 1600.0f));
  bn_finalize_kernel<<<(TT * 16 + 63) / 64, 64, 0, stream>>>(
      st1c, cg1, cbe1, sc1c, TT * 16, 16, 1.0f / (NB * 1600.0f));

  conv2_stats_kernel<5, 16><<<nImg, 256, 0, stream>>>(veg, vw1, vb1, sc1v, vw2,
                                                      vb2, st2v);
  conv2_stats_kernel<1, 8><<<nImg, 256, 0, stream>>>(cwsi, cw1, cb1, sc1c, cw2,
                                                     cb2, st2c);

  bn_finalize_kernel<<<(TT * 16 + 63) / 64, 64, 0, stream>>>(
      st2v, vg2, vbe2, sc2v, TT * 16, 16, 1.0f / (NB * 400.0f));
  bn_finalize_kernel<<<(TT * 8 + 63) / 64, 64, 0, stream>>>(
      st2c, cg2, cbe2, sc2c, TT * 8, 8, 1.0f / (NB * 400.0f));

  features_kernel<5, 16><<<nImg, 256, 0, stream>>>(veg, vw1, vb1, sc1v, vw2,
                                                   vb2, sc2v, feats, 0);
  features_kernel<1, 8><<<nImg, 256, 0, stream>>>(cwsi, cw1, cb1, sc1c, cw2,
                                                  cb2, sc2c, feats, 400);
  emb_kernel<<<(NB * TT * 16 + 255) / 256, 256, 0, stream>>>(irr, emb, feats);

  // ---- SAGE layer 1 ----
  {
    int n = TT * NB * FEAT;
    fill_u32_kernel<<<(n + 255) / 256, 256, 0, stream>>>(agg1, ENC_NEG_INF, n);
  }
  agg_scatter_kernel<<<TT * EE, 256, 0, stream>>>(feats, src, dst, agg1, FEAT);
  sage_gemm_kernel<<<(TT * NB) / 32, 256, 0, stream>>>(agg1, feats, s1_wl,
                                                       s1_bl, s1_wr, h1, FEAT,
                                                       64, 4);
  // ---- SAGE layer 2 ----
  {
    int n = TT * NB * 64;
    fill_u32_kernel<<<(n + 255) / 256, 256, 0, stream>>>(agg2, ENC_NEG_INF, n);
  }
  agg_scatter_kernel<<<TT * EE, 256, 0, stream>>>(h1, src, dst, agg2, 64);
  sage_gemm_kernel<<<(TT * NB) / 64, 256, 0, stream>>>(agg2, h1, s2_wl, s2_bl,
                                                       s2_wr, h2, 64, 32, 2);

  // ---- LSTM over T steps ----
  for (int t = 0; t < TT; ++t) {
    lstm_step_kernel<<<NB / 32, 256, 0, stream>>>(h2 + (size_t)t * NB * 32, wih,
                                                  whh, bih, bhh, hS, cS);
  }

  // ---- final FC ----
  fc_kernel<<<(NB + 255) / 256, 256, 0, stream>>>(hS, fcw, fcb, out);
}